// Sampler_25323127177408
// MI455X (gfx1250) — compile-verified
//
#include <hip/hip_runtime.h>

#define N_SEGMENTS 4096
#define BLK 256

typedef float f32x4 __attribute__((ext_vector_type(4)));
typedef int   i32x4 __attribute__((ext_vector_type(4)));

// ---------- order-preserving float<->uint key for atomic max ----------
__device__ __forceinline__ unsigned f32_order_key(float x) {
    int b = __float_as_int(x);
    return (unsigned)(b ^ ((b >> 31) | 0x80000000));
}
__device__ __forceinline__ float f32_from_key(unsigned k) {
    int b = (k & 0x80000000u) ? (int)(k ^ 0x80000000u) : (int)~k;
    return __int_as_float(b);
}

// ---------- LDS byte-offset of a generic pointer (for async-to-LDS asm) ----
typedef __attribute__((address_space(3))) const unsigned char lds_cbyte;
__device__ __forceinline__ unsigned lds_off_of(const void* p) {
    return (unsigned)(unsigned long long)(lds_cbyte*)p;
}

// ---------------- K0: reset segment tables (every call!) -------------------
__global__ void __launch_bounds__(BLK)
k0_init(unsigned* __restrict__ seg_max_key, float* __restrict__ seg_sum) {
    int i = blockIdx.x * BLK + threadIdx.x;
    if (i < N_SEGMENTS) { seg_max_key[i] = 0u; seg_sum[i] = 0.0f; }
}

// ------- K1: y = logits[edge_id] + noise; segmented max (sorted ids) -------
// 4 elements/thread, b128 streams; NT hints on read-once streams so the
// 80MB logits table + 32MB y + 32MB seg_ids stay resident in the 192MB L2.
__global__ void __launch_bounds__(BLK)
k1_gather_max(const float* __restrict__ logits,
              const float* __restrict__ loglog_u,
              const int*   __restrict__ edge_id,
              const int*   __restrict__ seg_ids,
              float*       __restrict__ y_out,
              unsigned*    __restrict__ seg_max_key,
              int n) {
    int i0   = (blockIdx.x * BLK + threadIdx.x) * 4;
    int lane = threadIdx.x & 31;

    int   segs[4];
    float vals[4];
    if (i0 + 3 < n) {
        i32x4 e4 = __builtin_nontemporal_load((const i32x4*)(edge_id  + i0));
        f32x4 u4 = __builtin_nontemporal_load((const f32x4*)(loglog_u + i0));
        i32x4 s4 = *(const i32x4*)(seg_ids + i0);            // reused later: RT
        f32x4 y4;
        y4.x = logits[e4.x] + u4.x;  y4.y = logits[e4.y] + u4.y;
        y4.z = logits[e4.z] + u4.z;  y4.w = logits[e4.w] + u4.w;
        *(f32x4*)(y_out + i0) = y4;                          // reused in K2/K3: RT
        segs[0] = s4.x; segs[1] = s4.y; segs[2] = s4.z; segs[3] = s4.w;
        vals[0] = y4.x; vals[1] = y4.y; vals[2] = y4.z; vals[3] = y4.w;
    } else {
        #pragma unroll
        for (int k = 0; k < 4; ++k) {
            int i = i0 + k;
            if (i < n) {
                float yv = logits[edge_id[i]] + loglog_u[i];
                y_out[i] = yv;
                segs[k] = seg_ids[i]; vals[k] = yv;
            } else { segs[k] = -1; vals[k] = -3.0e38f; }
        }
    }

    // local run merge (sorted ids): only head/tail runs can cross threads
    float run = vals[0]; int rs = segs[0];
    int head_seg = segs[0]; float head_val = 0.0f; bool have_head = false;
    #pragma unroll
    for (int k = 1; k < 4; ++k) {
        if (segs[k] == rs) { run = fmaxf(run, vals[k]); }
        else {
            if (!have_head) { head_val = run; have_head = true; }
            else if (rs >= 0) atomicMax(&seg_max_key[rs], f32_order_key(run)); // internal run
            rs = segs[k]; run = vals[k];
        }
    }

    // wave32 segmented scan over (tail_seg, tail_val)
    float sv = run; int tseg = rs;
    #pragma unroll
    for (int off = 1; off < 32; off <<= 1) {
        float ov = __shfl_up(sv,   off, 32);
        int   os = __shfl_up(tseg, off, 32);
        if (lane >= off && os == tseg) sv = fmaxf(sv, ov);
    }
    float prev_sv   = __shfl_up(sv, 1, 32);
    int   prev_tseg = __shfl_up(tseg, 1, 32);
    int   next_hseg = __shfl_down(head_seg, 1, 32);

    if (have_head && head_seg >= 0) {          // head run ends inside this thread
        float hv = head_val;
        if (lane > 0 && prev_tseg == head_seg) hv = fmaxf(hv, prev_sv);
        atomicMax(&seg_max_key[head_seg], f32_order_key(hv));
    }
    bool tail_end = (lane == 31) || (next_hseg != tseg);
    if (tail_end && tseg >= 0)
        atomicMax(&seg_max_key[tseg], f32_order_key(sv));
}

// ------- K2: e = exp(y - segmax); segmented sum (sorted ids) ---------------
__global__ void __launch_bounds__(BLK)
k2_expsum(const float*    __restrict__ y,
          const int*      __restrict__ seg_ids,
          const unsigned* __restrict__ seg_max_key,
          float*          __restrict__ seg_sum,
          int n) {
    int i0   = (blockIdx.x * BLK + threadIdx.x) * 4;
    int lane = threadIdx.x & 31;

    int   segs[4];
    float vals[4];
    if (i0 + 3 < n) {
        f32x4 y4 = *(const f32x4*)(y + i0);
        i32x4 s4 = *(const i32x4*)(seg_ids + i0);
        segs[0] = s4.x; segs[1] = s4.y; segs[2] = s4.z; segs[3] = s4.w;
        float m0 = f32_from_key(seg_max_key[s4.x]);
        float m1 = f32_from_key(seg_max_key[s4.y]);
        float m2 = f32_from_key(seg_max_key[s4.z]);
        float m3 = f32_from_key(seg_max_key[s4.w]);
        vals[0] = __expf(y4.x - m0); vals[1] = __expf(y4.y - m1);
        vals[2] = __expf(y4.z - m2); vals[3] = __expf(y4.w - m3);
    } else {
        #pragma unroll
        for (int k = 0; k < 4; ++k) {
            int i = i0 + k;
            if (i < n) {
                int s = seg_ids[i];
                segs[k] = s;
                vals[k] = __expf(y[i] - f32_from_key(seg_max_key[s]));
            } else { segs[k] = -1; vals[k] = 0.0f; }
        }
    }

    float run = vals[0]; int rs = segs[0];
    int head_seg = segs[0]; float head_val = 0.0f; bool have_head = false;
    #pragma unroll
    for (int k = 1; k < 4; ++k) {
        if (segs[k] == rs) { run += vals[k]; }
        else {
            if (!have_head) { head_val = run; have_head = true; }
            else if (rs >= 0) atomicAdd(&seg_sum[rs], run);      // internal run
            rs = segs[k]; run = vals[k];
        }
    }

    float sv = run; int tseg = rs;
    #pragma unroll
    for (int off = 1; off < 32; off <<= 1) {
        float ov = __shfl_up(sv,   off, 32);
        int   os = __shfl_up(tseg, off, 32);
        if (lane >= off && os == tseg) sv += ov;
    }
    float prev_sv   = __shfl_up(sv, 1, 32);
    int   prev_tseg = __shfl_up(tseg, 1, 32);
    int   next_hseg = __shfl_down(head_seg, 1, 32);

    if (have_head && head_seg >= 0) {
        float hv = head_val;
        if (lane > 0 && prev_tseg == head_seg) hv += prev_sv;
        atomicAdd(&seg_sum[head_seg], hv);
    }
    bool tail_end = (lane == 31) || (next_hseg != tseg);
    if (tail_end && tseg >= 0)
        atomicAdd(&seg_sum[tseg], sv);
}

// ------- K3: resolve sampled outputs; tables async-preloaded to LDS --------
__global__ void __launch_bounds__(BLK)
k3_output(const float*    __restrict__ y,
          const int*      __restrict__ seg_ids,
          const int*      __restrict__ ca_idx,
          const unsigned* __restrict__ seg_max_key,
          const float*    __restrict__ seg_sum,
          float*          __restrict__ out,
          int n) {
    __shared__ unsigned s_key[N_SEGMENTS];   // 16 KB
    __shared__ float    s_sum[N_SEGMENTS];   // 16 KB  (32 KB/block, 320 KB/WGP)

    // Async DMA both 4096-entry tables into LDS: 256 threads x 4 x b128 each.
    int base = threadIdx.x * 16;             // dword index; covers [0,4096) exactly
    #pragma unroll
    for (int q = 0; q < 4; ++q) {
        int idx = base + q * 4;
        unsigned           lk = lds_off_of(&s_key[idx]);
        unsigned long long gk = (unsigned long long)(const void*)(seg_max_key + idx);
        asm volatile("global_load_async_to_lds_b128 %0, %1, off"
                     :: "v"(lk), "v"(gk) : "memory");
        unsigned           ls = lds_off_of(&s_sum[idx]);
        unsigned long long gs = (unsigned long long)(const void*)(seg_sum + idx);
        asm volatile("global_load_async_to_lds_b128 %0, %1, off"
                     :: "v"(ls), "v"(gs) : "memory");
    }
    asm volatile("s_wait_asynccnt 0" ::: "memory");
    __syncthreads();

    int j0 = (blockIdx.x * BLK + threadIdx.x) * 4;
    if (j0 + 3 < n) {
        __builtin_prefetch(&ca_idx[j0 + 8192], 0, 0);            // global_prefetch_b8
        i32x4 c4 = __builtin_nontemporal_load((const i32x4*)(ca_idx + j0));
        f32x4 o4;
        {
            float yc = y[c4.x]; int s = seg_ids[c4.x];
            float ys = __expf(yc - f32_from_key(s_key[s])) / s_sum[s];
            o4.x = (1.0f - ys) + ys;
        }
        {
            float yc = y[c4.y]; int s = seg_ids[c4.y];
            float ys = __expf(yc - f32_from_key(s_key[s])) / s_sum[s];
            o4.y = (1.0f - ys) + ys;
        }
        {
            float yc = y[c4.z]; int s = seg_ids[c4.z];
            float ys = __expf(yc - f32_from_key(s_key[s])) / s_sum[s];
            o4.z = (1.0f - ys) + ys;
        }
        {
            float yc = y[c4.w]; int s = seg_ids[c4.w];
            float ys = __expf(yc - f32_from_key(s_key[s])) / s_sum[s];
            o4.w = (1.0f - ys) + ys;
        }
        __builtin_nontemporal_store(o4, (f32x4*)(out + j0));
    } else {
        for (int k = 0; k < 4; ++k) {
            int j = j0 + k;
            if (j < n) {
                int   c  = ca_idx[j];
                float yc = y[c];
                int   s  = seg_ids[c];
                float ys = __expf(yc - f32_from_key(s_key[s])) / s_sum[s];
                out[j]   = (1.0f - ys) + ys;
            }
        }
    }
}

// ---------------------------------------------------------------------------
extern "C" void kernel_launch(void* const* d_in, const int* in_sizes, int n_in,
                              void* d_out, int out_size, void* d_ws, size_t ws_size,
                              hipStream_t stream) {
    const float* edges_logits = (const float*)d_in[0];
    const float* loglog_u     = (const float*)d_in[1];
    const int*   edge_id      = (const int*)  d_in[2];
    const int*   seg_ids      = (const int*)  d_in[3];
    const int*   ca_idx       = (const int*)  d_in[4];
    float*       out          = (float*)      d_out;

    const int n_cand    = in_sizes[1];   // 8,000,000
    const int n_sampled = in_sizes[4];   // 1,000,000

    // Workspace: [ y : n_cand f32 ][ seg_max_key : 4096 u32 ][ seg_sum : 4096 f32 ]
    float*    ws_y        = (float*)d_ws;
    unsigned* seg_max_key = (unsigned*)((char*)d_ws + (size_t)n_cand * 4);
    float*    seg_sum     = (float*)((char*)seg_max_key + (size_t)N_SEGMENTS * 4);

    k0_init<<<(N_SEGMENTS + BLK - 1) / BLK, BLK, 0, stream>>>(seg_max_key, seg_sum);

    int tcand = (n_cand + 3) / 4;                 // 4 elems/thread
    int gcand = (tcand + BLK - 1) / BLK;
    k1_gather_max<<<gcand, BLK, 0, stream>>>(edges_logits, loglog_u, edge_id, seg_ids,
                                             ws_y, seg_max_key, n_cand);
    k2_expsum<<<gcand, BLK, 0, stream>>>(ws_y, seg_ids, seg_max_key, seg_sum, n_cand);

    int tsamp = (n_sampled + 3) / 4;
    int gsamp = (tsamp + BLK - 1) / BLK;
    k3_output<<<gsamp, BLK, 0, stream>>>(ws_y, seg_ids, ca_idx, seg_max_key, seg_sum,
                                         out, n_sampled);
}